// SelfMaskedAttention_59304908423850
// MI455X (gfx1250) — compile-verified
//
#include <hip/hip_runtime.h>

typedef float v2f  __attribute__((ext_vector_type(2)));
typedef float v8f  __attribute__((ext_vector_type(8)));
typedef unsigned int u32x4 __attribute__((ext_vector_type(4)));
typedef int   i32x4 __attribute__((ext_vector_type(4)));
typedef int   i32x8 __attribute__((ext_vector_type(8)));

// ---------------------------------------------------------------------------
// Tensor Data Mover: 2-D tile load Global -> LDS with optional LDS row pad.
// D# per CDNA5 ISA ch.8: group0 = {ctl, lds_addr, gaddr_lo, gaddr_hi|type},
// group1 = {mask/size/pad, dims, tile dims, dim0 stride}. Groups 2/3 unused
// (2-D). pad_int code: 0=2,1=4,2=8,3=16,4=32,5=64,6=128 DWORDs between pads;
// pad_amt code: N -> N+1 DWORDs inserted.
// This toolchain exposes the 6-arg builtin form:
//   (u32x4 g0, i32x8 g1, i32x4 g2, i32x4 g3, i32x8 aux, i32 cpol)
// ---------------------------------------------------------------------------
__device__ __forceinline__ unsigned lds_off(const void* p) {
    return (unsigned)(unsigned long long)p;   // LDS aperture: addr[31:0] = LDS byte offset
}

__device__ __forceinline__ void tdm_load_2d(const void* gaddr, unsigned ldsaddr,
                                            unsigned tile_w, unsigned tile_h,
                                            unsigned long long row_stride_elems,
                                            unsigned pad_int, unsigned pad_amt)
{
    unsigned long long ga = (unsigned long long)gaddr;
    u32x4 g0;
    g0[0] = 1u;                                    // count=1, user descriptor
    g0[1] = ldsaddr;                               // lds_addr [63:32]
    g0[2] = (unsigned)(ga & 0xffffffffu);          // global_addr lo
    g0[3] = (unsigned)((ga >> 32) & 0x01ffffffu)   // global_addr hi (bits 120:96)
          | (2u << 30);                            // type = 2 ("image")
    i32x8 g1;
    g1[0] = (int)((2u << 16)                       // data_size = 4 bytes
          | (1u << 20)                             // pad_enable
          | (pad_int << 22) | (pad_amt << 25));
    g1[1] = (int)((tile_w & 0xffffu) << 16);       // tensor_dim0[15:0] (= tile_w)
    g1[2] = (int)(((tile_w >> 16) & 0xffffu)       // tensor_dim0[31:16]
          | ((tile_h & 0xffffu) << 16));           // tensor_dim1[15:0] (= tile_h)
    g1[3] = (int)(((tile_h >> 16) & 0xffffu)
          | ((tile_w & 0xffffu) << 16));           // tile_dim0
    g1[4] = (int)(tile_h & 0xffffu);               // tile_dim1 (tile_dim2 = 0)
    g1[5] = (int)(row_stride_elems & 0xffffffffu); // tensor_dim0_stride lo
    g1[6] = (int)((row_stride_elems >> 32) & 0xffffu);
    g1[7] = 0;
    i32x4 z4 = {0, 0, 0, 0};
    i32x8 z8 = {0, 0, 0, 0, 0, 0, 0, 0};
    __builtin_amdgcn_tensor_load_to_lds(g0, g1, z4, z4, z8, 0);
}

#define BM 128
#define BN 128
#define BK 32
#define BKP 33    // TDM pad: 32 + 1  (pad_int=4, pad_amt=0)
#define BNP 132   // TDM pad: 128 + 4 (pad_int=6, pad_amt=3)
#define KVP 68    // TDM pad: 64 + 4  (pad_int=5, pad_amt=3)
#define PSTR 132

// ---------------------------------------------------------------------------
// C[M,N] = A[M,K] @ B[K,N] + bias[N]   (fp32, V_WMMA_F32_16X16X4_F32)
// 8 waves: wave grid 4(M) x 2(N), each 32x64 = 2x4 16x16 tiles.
// TDM double-buffered staging, synced with s_wait_tensorcnt.
// ---------------------------------------------------------------------------
__global__ __launch_bounds__(256) void gemm_bias_kernel(
    const float* __restrict__ A, const float* __restrict__ Bw,
    const float* __restrict__ bias, float* __restrict__ C,
    int M, int N, int K)
{
    extern __shared__ float smem[];
    float* As = smem;                 // 2 x BM*BKP
    float* Bs = smem + 2 * BM * BKP;  // 2 x BK*BNP

    const int tid  = threadIdx.x;
    const int lane = tid & 31;
    const int wave = tid >> 5;
    const int wm   = wave & 3;
    const int wn   = wave >> 2;
    const int m0   = blockIdx.y * BM;
    const int n0   = blockIdx.x * BN;

    const int r    = lane & 15;
    const int hi   = lane >> 4;
    const int ksel = hi * 2;

    if (wave == 0) {   // prologue DMA, stage 0
        tdm_load_2d(A + (size_t)m0 * K, lds_off(As), BK, BM, (unsigned)K, 4, 0);
        tdm_load_2d(Bw + n0,            lds_off(Bs), BN, BK, (unsigned)N, 6, 3);
    }

    v8f acc[2][4] = {};

    for (int k0 = 0; k0 < K; k0 += BK) {
        const int buf = (k0 / BK) & 1;
        if (wave == 0) {
            if (k0 + BK < K) {   // stream stage s+1 into alternate buffer
                const int nb = buf ^ 1;
                tdm_load_2d(A + (size_t)m0 * K + (k0 + BK),
                            lds_off(As + nb * BM * BKP), BK, BM, (unsigned)K, 4, 0);
                tdm_load_2d(Bw + (size_t)(k0 + BK) * N + n0,
                            lds_off(Bs + nb * BK * BNP), BN, BK, (unsigned)N, 6, 3);
                __builtin_amdgcn_s_wait_tensorcnt(2);   // stage s complete
            } else {
                __builtin_amdgcn_s_wait_tensorcnt(0);
            }
        }
        __syncthreads();

        if (k0 + 2 * BK < K) {  // pull slab-after-next toward L2
            __builtin_prefetch(A + (size_t)(m0 + (tid >> 1)) * K + k0 + 2 * BK, 0, 0);
            __builtin_prefetch(Bw + (size_t)(k0 + 2 * BK + (tid >> 3)) * N + n0 + (tid & 7) * 16, 0, 0);
        }

        const float* Ab = As + buf * BM * BKP;
        const float* Bb = Bs + buf * BK * BNP;

        #pragma unroll
        for (int kk = 0; kk < BK; kk += 4) {
            v2f afrag[2], bfrag[4];
            #pragma unroll
            for (int i = 0; i < 2; ++i) {
                int row = wm * 32 + i * 16 + r;
                afrag[i].x = Ab[row * BKP + kk + ksel];
                afrag[i].y = Ab[row * BKP + kk + ksel + 1];
            }
            #pragma unroll
            for (int j = 0; j < 4; ++j) {
                int col = wn * 64 + j * 16 + r;
                bfrag[j].x = Bb[(kk + ksel) * BNP + col];
                bfrag[j].y = Bb[(kk + ksel + 1) * BNP + col];
            }
            #pragma unroll
            for (int i = 0; i < 2; ++i)
                #pragma unroll
                for (int j = 0; j < 4; ++j)
                    acc[i][j] = __builtin_amdgcn_wmma_f32_16x16x4_f32(
                        false, afrag[i], false, bfrag[j],
                        (short)0, acc[i][j], false, false);
        }
        __syncthreads();   // all waves done with buf before it is re-streamed
    }

    #pragma unroll
    for (int i = 0; i < 2; ++i) {
        int mrow = m0 + wm * 32 + i * 16 + hi * 8;
        #pragma unroll
        for (int j = 0; j < 4; ++j) {
            int col = n0 + wn * 64 + j * 16 + r;
            float b = bias[col];
            #pragma unroll
            for (int e = 0; e < 8; ++e)
                C[(size_t)(mrow + e) * N + col] = acc[i][j][e] + b;
        }
    }
}

// ---------------------------------------------------------------------------
// Fused attention. Per block: one (batch,head) x 128-query tile.
// Reference semantics: scores *= tril mask (zeros, NOT -inf) BEFORE
// scale+softmax -> masked entries contribute exp(0-m); visit ALL key blocks.
// K stored naturally [key][d] (stride 68): the B-fragment "transposed" read
// Ks[(t*16+r)*68 + k] is bank-conflict-free (68 = 4 mod 64).
// Each wave owns 16 complete query rows -> softmax via shfl_xor in 16-lane
// halves; P staged in a private LDS region (no cross-wave sharing).
// K/V double-buffered via TDM + s_wait_tensorcnt.
// ---------------------------------------------------------------------------
__global__ __launch_bounds__(256) void attn_kernel(
    const float* __restrict__ qkv, float* __restrict__ ctx)
{
    extern __shared__ float smem[];
    float* Qs = smem;                   // 128 x KVP
    float* Ks = Qs + 128 * KVP;         // 2 x (128 x KVP)
    float* Vs = Ks + 2 * 128 * KVP;     // 2 x (128 x KVP)
    float* Ps = Vs + 2 * 128 * KVP;     // 128 x PSTR

    const int tid  = threadIdx.x;
    const int lane = tid & 31;
    const int wave = tid >> 5;
    const int r    = lane & 15;
    const int hi   = lane >> 4;
    const int ksel = hi * 2;

    const int q0 = blockIdx.x * 128;
    const int bh = blockIdx.y;
    const int bi = bh >> 4;
    const int h  = bh & 15;

    const size_t RS = 3072;
    const float* Qg = qkv + (size_t)(bi * 2048 + q0) * RS + h * 64;
    const float* Kg = qkv + (size_t)(bi * 2048) * RS + 1024 + h * 64;
    const float* Vg = Kg + 1024;

    if (wave == 0) {   // prologue DMA: Q once, K0/V0
        tdm_load_2d(Qg, lds_off(Qs), 64, 128, RS, 5, 3);
        tdm_load_2d(Kg, lds_off(Ks), 64, 128, RS, 5, 3);
        tdm_load_2d(Vg, lds_off(Vs), 64, 128, RS, 5, 3);
    }

    v8f o[4] = {};
    float mrow[8], lrow[8];
    #pragma unroll
    for (int i = 0; i < 8; ++i) { mrow[i] = -3.0e38f; lrow[i] = 0.0f; }
    const float scale = 0.125f;   // 1/sqrt(64)

    for (int j0 = 0; j0 < 2048; j0 += 128) {
        const int buf = (j0 >> 7) & 1;
        if (wave == 0) {
            if (j0 + 128 < 2048) {   // stream next K/V block
                const int nb = buf ^ 1;
                tdm_load_2d(Kg + (size_t)(j0 + 128) * RS, lds_off(Ks + nb * 128 * KVP), 64, 128, RS, 5, 3);
                tdm_load_2d(Vg + (size_t)(j0 + 128) * RS, lds_off(Vs + nb * 128 * KVP), 64, 128, RS, 5, 3);
                __builtin_amdgcn_s_wait_tensorcnt(2);   // current block resident
            } else {
                __builtin_amdgcn_s_wait_tensorcnt(0);
            }
        }
        __syncthreads();

        const float* Kb = Ks + buf * 128 * KVP;
        const float* Vb = Vs + buf * 128 * KVP;

        // ---- S = Q @ K^T : wave rows wave*16..+15, all 128 keys ----
        v8f s[8] = {};
        #pragma unroll
        for (int kk = 0; kk < 64; kk += 4) {
            v2f a;
            int qr = wave * 16 + r;
            a.x = Qs[qr * KVP + kk + ksel];
            a.y = Qs[qr * KVP + kk + ksel + 1];
            #pragma unroll
            for (int t = 0; t < 8; ++t) {
                v2f b;   // B[k][n] = K[key=n][d=k] read from natural layout
                int key = t * 16 + r;
                b.x = Kb[key * KVP + kk + ksel];
                b.y = Kb[key * KVP + kk + ksel + 1];
                s[t] = __builtin_amdgcn_wmma_f32_16x16x4_f32(
                    false, a, false, b, (short)0, s[t], false, false);
            }
        }

        // ---- mask*scale + online softmax ----
        const int qrb = q0 + wave * 16 + hi * 8;
        float pm[8];
        #pragma unroll
        for (int i = 0; i < 8; ++i) pm[i] = -3.0e38f;
        #pragma unroll
        for (int t = 0; t < 8; ++t) {
            int kcol = j0 + t * 16 + r;
            #pragma unroll
            for (int i = 0; i < 8; ++i) {
                float v = (kcol <= qrb + i) ? s[t][i] * scale : 0.0f;
                s[t][i] = v;
                pm[i] = fmaxf(pm[i], v);
            }
        }
        #pragma unroll
        for (int i = 0; i < 8; ++i) {
            float v = pm[i];
            v = fmaxf(v, __shfl_xor(v, 1, 32));
            v = fmaxf(v, __shfl_xor(v, 2, 32));
            v = fmaxf(v, __shfl_xor(v, 4, 32));
            v = fmaxf(v, __shfl_xor(v, 8, 32));
            pm[i] = v;
        }
        float corr[8], ps[8];
        #pragma unroll
        for (int i = 0; i < 8; ++i) {
            float mn = fmaxf(mrow[i], pm[i]);
            corr[i] = expf(mrow[i] - mn);
            mrow[i] = mn;
            ps[i] = 0.0f;
        }
        #pragma unroll
        for (int t = 0; t < 8; ++t)
            #pragma unroll
            for (int i = 0; i < 8; ++i) {
                float p = expf(s[t][i] - mrow[i]);
                s[t][i] = p;
                ps[i] += p;
            }
        #pragma unroll
        for (int i = 0; i < 8; ++i) {
            float v = ps[i];
            v += __shfl_xor(v, 1, 32);
            v += __shfl_xor(v, 2, 32);
            v += __shfl_xor(v, 4, 32);
            v += __shfl_xor(v, 8, 32);
            lrow[i] = lrow[i] * corr[i] + v;
        }
        #pragma unroll
        for (int t = 0; t < 4; ++t)
            #pragma unroll
            for (int i = 0; i < 8; ++i)
                o[t][i] *= corr[i];

        // ---- P: C-layout -> A-layout via wave-private LDS rows ----
        #pragma unroll
        for (int t = 0; t < 8; ++t)
            #pragma unroll
            for (int i = 0; i < 8; ++i)
                Ps[(wave * 16 + hi * 8 + i) * PSTR + t * 16 + r] = s[t][i];
        // (no barrier: each wave reads back only its own 16 rows; DS in-order)

        // ---- O += P @ V ----
        #pragma unroll
        for (int kk = 0; kk < 128; kk += 4) {
            v2f a;
            int pr = wave * 16 + r;
            a.x = Ps[pr * PSTR + kk + ksel];
            a.y = Ps[pr * PSTR + kk + ksel + 1];
            #pragma unroll
            for (int t = 0; t < 4; ++t) {
                v2f b;
                b.x = Vb[(kk + ksel) * KVP + t * 16 + r];
                b.y = Vb[(kk + ksel + 1) * KVP + t * 16 + r];
                o[t] = __builtin_amdgcn_wmma_f32_16x16x4_f32(
                    false, a, false, b, (short)0, o[t], false, false);
            }
        }
        __syncthreads();   // all waves done with buf before it is re-streamed
    }

    #pragma unroll
    for (int i = 0; i < 8; ++i) lrow[i] = 1.0f / lrow[i];
    const int orb = q0 + wave * 16 + hi * 8;
    #pragma unroll
    for (int t = 0; t < 4; ++t)
        #pragma unroll
        for (int i = 0; i < 8; ++i)
            ctx[((size_t)(bi * 2048) + orb + i) * 1024 + h * 64 + t * 16 + r] =
                o[t][i] * lrow[i];
}

// ---------------------------------------------------------------------------
extern "C" void kernel_launch(void* const* d_in, const int* in_sizes, int n_in,
                              void* d_out, int out_size, void* d_ws, size_t ws_size,
                              hipStream_t stream)
{
    const float* x    = (const float*)d_in[0];   // (4,2048,1024)
    const float* Wqkv = (const float*)d_in[1];   // (1024,3072)
    const float* bqkv = (const float*)d_in[2];   // (3072,)
    const float* Wout = (const float*)d_in[3];   // (1024,1024)
    const float* bout = (const float*)d_in[4];   // (1024,)
    float* out = (float*)d_out;                  // (4,2048,1024)

    float* qkvbuf = (float*)d_ws;                     // 8192 x 3072
    float* ctxbuf = qkvbuf + (size_t)8192 * 3072;     // 8192 x 1024

    const size_t gemm_lds = (size_t)(2 * BM * BKP + 2 * BK * BNP) * sizeof(float); // 67.6 KB
    const size_t attn_lds = (size_t)(5 * 128 * KVP + 128 * PSTR) * sizeof(float);  // 241.7 KB (<320 KB/WGP)

    (void)hipFuncSetAttribute((const void*)gemm_bias_kernel,
                              hipFuncAttributeMaxDynamicSharedMemorySize, (int)gemm_lds);
    (void)hipFuncSetAttribute((const void*)attn_kernel,
                              hipFuncAttributeMaxDynamicSharedMemorySize, (int)attn_lds);

    dim3 blk(256);
    gemm_bias_kernel<<<dim3(3072 / BN, 8192 / BM), blk, gemm_lds, stream>>>(
        x, Wqkv, bqkv, qkvbuf, 8192, 3072, 1024);

    attn_kernel<<<dim3(16, 64), blk, attn_lds, stream>>>(qkvbuf, ctxbuf);

    gemm_bias_kernel<<<dim3(1024 / BN, 8192 / BM), blk, gemm_lds, stream>>>(
        ctxbuf, Wout, bout, out, 8192, 1024, 1024);
}